// StableSpikingCoreFlow_12558484374115
// MI455X (gfx1250) — compile-verified
//
#include <hip/hip_runtime.h>

typedef __attribute__((ext_vector_type(2))) float v2f;
typedef __attribute__((ext_vector_type(4))) float v4f;
typedef __attribute__((ext_vector_type(8))) float v8f;

#define SSCF_B   128
#define SSCF_D   2048
#define SSCF_N   2048
#define SSCF_T   64
#define SSCF_L   4
#define SSCF_OUT 1024
#define SSCF_KS  4                    // K split factor (partial-sum buffers)
#define SSCF_KC  (SSCF_D / SSCF_KS)   // 512 K per block

// ---------------------------------------------------------------------------
// g[b][a] = src[b][axon[a]]   (feature gather, row length 2048 for all layers)
// ---------------------------------------------------------------------------
__global__ void sscf_gather(const float* __restrict__ src,
                            const int* __restrict__ axon,
                            float* __restrict__ g) {
    int tid = blockIdx.x * blockDim.x + threadIdx.x;   // 0 .. B*D-1
    int b = tid >> 11;
    int a = tid & (SSCF_D - 1);
    g[tid] = src[(b << 11) + axon[a]];
}

// ---------------------------------------------------------------------------
// Partial GEMM: part[kc][b][n] = sum_{a in chunk kc} W[n][a] * g[b][a]
// Orientation: WMMA D(16x16) = A(W rows, 16xK) x B(g^T, Kx16 batch cols).
// blockIdx.x = n-tile (16 rows of W), blockIdx.y = k-chunk, wave = b-tile.
// Two independent accumulators break the WMMA RAW dependency chain.
// ---------------------------------------------------------------------------
__global__ __launch_bounds__(256)
void sscf_gemm_partial(const float* __restrict__ Wl,   // (N, D) row-major
                       const float* __restrict__ g,    // (B, D) row-major
                       float* __restrict__ part) {     // (KS, B, N)
    const int wave = threadIdx.x >> 5;   // 0..7  -> batch tile
    const int lane = threadIdx.x & 31;
    const int l16  = lane & 15;
    const int hi   = lane >> 4;          // 0: K=0,1   1: K=2,3
    const int m_base = blockIdx.x * 16;  // neuron rows
    const int k0     = blockIdx.y * SSCF_KC;
    const int b_base = wave * 16;

    const float* Arow = Wl + (size_t)(m_base + l16) * SSCF_D + k0 + 2 * hi;
    const float* Brow = g  + (size_t)(b_base + l16) * SSCF_D + k0 + 2 * hi;

    v8f acc0 = {};
    v8f acc1 = {};
    for (int kb = 0; kb < SSCF_KC; kb += 32) {
        // stream-prefetch W a little ahead, staying inside this k-chunk
        if (kb + 288 <= SSCF_KC)
            __builtin_prefetch(Arow + kb + 256, 0, 1);
#pragma unroll
        for (int kk = 0; kk < 32; kk += 8) {
            v2f a0 = *(const v2f*)(Arow + kb + kk);       // A: 16x4 f32 fragment
            v2f b0 = *(const v2f*)(Brow + kb + kk);       // B: 4x16 f32 fragment
            acc0 = __builtin_amdgcn_wmma_f32_16x16x4_f32(
                false, a0, false, b0, (short)0, acc0, false, false);
            v2f a1 = *(const v2f*)(Arow + kb + kk + 4);
            v2f b1 = *(const v2f*)(Brow + kb + kk + 4);
            acc1 = __builtin_amdgcn_wmma_f32_16x16x4_f32(
                false, a1, false, b1, (short)0, acc1, false, false);
        }
    }

    v8f acc = acc0 + acc1;   // fixed-order combine: deterministic

    // D layout: acc[v] -> row n = m_base + v + 8*hi, col b = b_base + l16
    float* dst = part + (size_t)blockIdx.y * (SSCF_B * SSCF_N)
                      + (size_t)(b_base + l16) * SSCF_N + m_base + 8 * hi;
    v4f lo4 = {acc[0], acc[1], acc[2], acc[3]};
    v4f hi4 = {acc[4], acc[5], acc[6], acc[7]};
    *(v4f*)dst       = lo4;
    *((v4f*)dst + 1) = hi4;
}

// ---------------------------------------------------------------------------
// Reduce K-partials (fixed order: deterministic), run the exact 64-step
// integrate-and-fire scan, write cnt*scale (scale = 1/T for hidden layers,
// 1.0 for the last layer where the output needs the raw spike count).
// ---------------------------------------------------------------------------
__global__ void sscf_spike(const float* __restrict__ part,
                           const float* __restrict__ thr_ptr,
                           float* __restrict__ r, float scale) {
    int tid = blockIdx.x * blockDim.x + threadIdx.x;   // 0 .. B*N-1
    const float thr = *thr_ptr;
    float a = part[tid];
#pragma unroll
    for (int kc = 1; kc < SSCF_KS; ++kc)
        a += part[(size_t)kc * (SSCF_B * SSCF_N) + tid];

    float m = 0.0f, cnt = 0.0f;
    for (int t = 0; t < SSCF_T; ++t) {
        m += a;
        float s = (m > thr) ? 1.0f : 0.0f;  // matches reference: s=(m>thr); m-=thr*s
        m -= thr * s;
        cnt += s;
    }
    r[tid] = cnt * scale;
}

// ---------------------------------------------------------------------------
// out[b][o] = cnt[b][out_idx[o]]
// ---------------------------------------------------------------------------
__global__ void sscf_out_gather(const float* __restrict__ cnt,
                                const int* __restrict__ oidx,
                                float* __restrict__ out) {
    int tid = blockIdx.x * blockDim.x + threadIdx.x;   // 0 .. B*OUT-1
    int b = tid >> 10;
    int o = tid & (SSCF_OUT - 1);
    out[tid] = cnt[(b << 11) + oidx[o]];
}

extern "C" void kernel_launch(void* const* d_in, const int* in_sizes, int n_in,
                              void* d_out, int out_size, void* d_ws, size_t ws_size,
                              hipStream_t stream) {
    (void)in_sizes; (void)n_in; (void)out_size; (void)ws_size;
    const float* x    = (const float*)d_in[0];                 // (B, D)
    const float* W    = (const float*)d_in[1];                 // (L, N, D)
    const float* thr  = (const float*)d_in[2];                 // (L,)
    const int*   axon = (const int*)d_in[3];                   // (L, D)
    const int*   oidx = (const int*)d_in[4];                   // (OUT,)
    // d_in[5] = simulation_length (fixed T=64 in reference)
    float* out = (float*)d_out;                                // (B, OUT)

    float* gbuf = (float*)d_ws;                                // B*D
    float* rbuf = gbuf + (size_t)SSCF_B * SSCF_D;              // B*N
    float* part = rbuf + (size_t)SSCF_B * SSCF_N;              // KS*B*N

    const float* src = x;
    for (int l = 0; l < SSCF_L; ++l) {
        sscf_gather<<<(SSCF_B * SSCF_D) / 256, 256, 0, stream>>>(
            src, axon + l * SSCF_D, gbuf);

        dim3 ggrid(SSCF_N / 16, SSCF_KS);
        sscf_gemm_partial<<<ggrid, 256, 0, stream>>>(
            W + (size_t)l * SSCF_N * SSCF_D, gbuf, part);

        float scale = (l == SSCF_L - 1) ? 1.0f : (1.0f / (float)SSCF_T);
        sscf_spike<<<(SSCF_B * SSCF_N) / 256, 256, 0, stream>>>(
            part, thr + l, rbuf, scale);

        src = rbuf;
    }
    sscf_out_gather<<<(SSCF_B * SSCF_OUT) / 256, 256, 0, stream>>>(
        rbuf, oidx, out);
}